// GaussianSurfaceRegularizer_24927990186061
// MI455X (gfx1250) — compile-verified
//
#include <hip/hip_runtime.h>
#include <hip/hip_bf16.h>

typedef __attribute__((ext_vector_type(2))) float v2f;
typedef __attribute__((ext_vector_type(8))) float v8f;

#define N_PTS   8192
#define TILE    16
#define NTILES  (N_PTS / TILE)   // 512 column tiles
#define CSPLIT  16               // column-range splits per row tile (8192 waves total)

// ws layout (floats): [0]=flat_sum  [1]=align_sum  [2]=align_cnt  [3]=dens_sum

__device__ inline float wave_sum32(float v) {
#pragma unroll
  for (int off = 16; off >= 1; off >>= 1) v += __shfl_xor(v, off, 32);
  return v;
}

__global__ void init_ws_kernel(float* ws) {
  if (threadIdx.x < 4) ws[threadIdx.x] = 0.0f;
}

// ---------------- flatness: O(N) elementwise + reduce ----------------
__global__ void flatness_kernel(const float* __restrict__ scales,
                                float* __restrict__ ws) {
  int i = blockIdx.x * blockDim.x + threadIdx.x;
  float local = 0.0f;
  if (i < N_PTS) {
    float a = expf(scales[i * 3 + 0]);
    float b = expf(scales[i * 3 + 1]);
    float c = expf(scales[i * 3 + 2]);
    float lo  = fminf(a, fminf(b, c));
    float hi  = fmaxf(a, fmaxf(b, c));
    float mid = a + b + c - lo - hi;
    float aspect = hi / (lo + 1e-8f);
    float flat_reward = logf(aspect + 1e-8f);
    float disc_sim = 1.0f / (fabsf(hi - mid) + 0.001f);
    local = flat_reward + 0.1f * disc_sim;
  }
  local = wave_sum32(local);
  if ((threadIdx.x & 31) == 0) atomicAdd(ws + 0, local);
}

// ---------------- alignment: WMMA 16x16x4 f32 pairwise tiles ----------------
// One wave per (row tile, column split). A/B per-lane layout (f32 16x4):
//   lanes 0..15  hold K=0,1 for M(or N)=lane
//   lanes 16..31 hold K=2,3 for M(or N)=lane-16
// D layout: VGPR v, lane l -> row = v + 8*(l>=16), col = l&15.
__global__ __launch_bounds__(32)
void align_kernel(const float* __restrict__ pos,
                  const float* __restrict__ rot,
                  float* __restrict__ ws) {
  __shared__ float ssq[16];

  const int lane  = threadIdx.x;     // 0..31 (wave32)
  const int half  = lane >> 4;       // 0 or 1 (K half)
  const int lr    = lane & 15;       // row/col within tile
  const int rtile = blockIdx.x;
  const int rbase = rtile * TILE;
  const int kb    = half * 2;        // K base: 0 or 2

  // ---- A matrices for this wave's 16 rows ----
  const int arow = rbase + lr;
  float a0 = pos[arow * 3 + kb];
  float a1 = (kb + 1 < 3) ? pos[arow * 3 + kb + 1] : 0.0f;  // pad K=3 with 0
  v2f Ap = {a0, a1};
  v2f Ar = {rot[arow * 4 + kb], rot[arow * 4 + kb + 1]};    // quats: K=0..3 exact

  // row |p|^2: combine the two K-halves across the lane pair (l, l^16)
  float part = a0 * a0 + a1 * a1;
  float sq_full = part + __shfl_xor(part, 16, 32);
  if (half == 0) ssq[lr] = sq_full;
  __syncthreads();
  float sqr[8];
#pragma unroll
  for (int v = 0; v < 8; ++v) sqr[v] = ssq[v + half * 8];

  float asum = 0.0f, acnt = 0.0f;

  const int jt0 = blockIdx.y * (NTILES / CSPLIT);
  const int jt1 = jt0 + (NTILES / CSPLIT);
  for (int jt = jt0; jt < jt1; ++jt) {
    const int col = jt * TILE + lr;
    float b0 = pos[col * 3 + kb];
    float b1 = (kb + 1 < 3) ? pos[col * 3 + kb + 1] : 0.0f;
    v2f Bp = {b0, b1};
    v2f Br = {rot[col * 4 + kb], rot[col * 4 + kb + 1]};

    float cpart = b0 * b0 + b1 * b1;
    float sqc = cpart + __shfl_xor(cpart, 16, 32);   // column |p|^2

    v8f zero = {};
    // 16x16 tile of position dot products, K=4 (padded from 3)
    v8f Dp = __builtin_amdgcn_wmma_f32_16x16x4_f32(
        false, Ap, false, Bp, (short)0, zero, false, false);
    // 16x16 tile of rotation dot products, K=4 exact
    v8f Dq = __builtin_amdgcn_wmma_f32_16x16x4_f32(
        false, Ar, false, Br, (short)0, zero, false, false);

#pragma unroll
    for (int v = 0; v < 8; ++v) {
      float d2   = sqr[v] + sqc - 2.0f * Dp[v];
      float dist = sqrtf(fmaxf(d2, 0.0f) + 1e-12f);
      int rowg   = rbase + v + half * 8;
      bool nb    = (dist < 0.02f) && (rowg != col);
      if (nb) {
        asum += 1.0f - fabsf(Dq[v]);
        acnt += 1.0f;
      }
    }
  }

  asum = wave_sum32(asum);
  acnt = wave_sum32(acnt);
  if (lane == 0) {
    atomicAdd(ws + 1, asum);
    atomicAdd(ws + 2, acnt);
  }
}

// ---------------- density: one wave per row, 8-NN over N points ----------------
// Each lane scans a strided 1/32 slice keeping a sorted 9-list (self + 8 NN),
// then a 5-level shfl_xor butterfly merges the 32 sorted lists.
#define KNN_WAVES_PER_BLOCK 8
__global__ __launch_bounds__(KNN_WAVES_PER_BLOCK * 32)
void knn_kernel(const float* __restrict__ pos,
                const float* __restrict__ opac,
                float* __restrict__ ws) {
  __shared__ float spartial[KNN_WAVES_PER_BLOCK];

  const int lane = threadIdx.x & 31;
  const int wave = threadIdx.x >> 5;
  const int row  = blockIdx.x * KNN_WAVES_PER_BLOCK + wave;

  float px = pos[row * 3 + 0], py = pos[row * 3 + 1], pz = pos[row * 3 + 2];
  float sqi = px * px + py * py + pz * pz;

  float knn[9];  // ascending
#pragma unroll
  for (int t = 0; t < 9; ++t) knn[t] = 3.0e38f;

  for (int j = lane; j < N_PTS; j += 32) {           // coalesced strided scan
    float qx = pos[j * 3 + 0], qy = pos[j * 3 + 1], qz = pos[j * 3 + 2];
    float dot = px * qx + py * qy + pz * qz;
    float sqj = qx * qx + qy * qy + qz * qz;
    float d2  = sqi + sqj - 2.0f * dot;              // same formula as reference
    float dist = sqrtf(fmaxf(d2, 0.0f) + 1e-12f);
    if (dist < knn[8]) {
      knn[8] = dist;
#pragma unroll
      for (int t = 8; t > 0; --t) {                  // single bubble pass
        float lo = fminf(knn[t - 1], knn[t]);
        float hi = fmaxf(knn[t - 1], knn[t]);
        knn[t - 1] = lo;
        knn[t]     = hi;
      }
    }
  }

  // merge 32 sorted per-lane lists -> global 9 smallest (every lane ends equal)
#pragma unroll
  for (int off = 1; off < 32; off <<= 1) {
    float other[9];
#pragma unroll
    for (int t = 0; t < 9; ++t) other[t] = __shfl_xor(knn[t], off, 32);
#pragma unroll
    for (int t = 0; t < 9; ++t) {
      if (other[t] < knn[8]) {                       // other[] ascending: prunes fast
        knn[8] = other[t];
#pragma unroll
        for (int u = 8; u > 0; --u) {
          float lo = fminf(knn[u - 1], knn[u]);
          float hi = fmaxf(knn[u - 1], knn[u]);
          knn[u - 1] = lo;
          knn[u]     = hi;
        }
      }
    }
  }

  if (lane == 0) {
    float sum = 0.0f;
#pragma unroll
    for (int t = 1; t < 9; ++t) sum += knn[t];       // drop self (min), keep 8
    float avg = sum * 0.125f;
    spartial[wave] = fabsf(avg - 0.01f) * opac[row];
  }
  __syncthreads();
  if (threadIdx.x == 0) {
    float s = 0.0f;
#pragma unroll
    for (int w = 0; w < KNN_WAVES_PER_BLOCK; ++w) s += spartial[w];
    atomicAdd(ws + 3, s);
  }
}

// ---------------- finalize ----------------
__global__ void finalize_kernel(const float* __restrict__ ws,
                                float* __restrict__ out) {
  if (threadIdx.x == 0 && blockIdx.x == 0) {
    const float invN = 1.0f / (float)N_PTS;
    float flatness_loss = -(ws[0] * invN);
    float cnt = ws[2];
    float alignment_loss = (cnt > 0.0f) ? (ws[1] / fmaxf(cnt, 1.0f)) : 0.0f;
    float density_loss = ws[3] * invN;
    out[0] = 1.0f * flatness_loss + 0.5f * alignment_loss + 0.2f * density_loss;
  }
}

extern "C" void kernel_launch(void* const* d_in, const int* in_sizes, int n_in,
                              void* d_out, int out_size, void* d_ws, size_t ws_size,
                              hipStream_t stream) {
  const float* pos = (const float*)d_in[0];   // (N,3) f32
  const float* rot = (const float*)d_in[1];   // (N,4) f32
  const float* scl = (const float*)d_in[2];   // (N,3) f32
  const float* opa = (const float*)d_in[3];   // (N,1) f32
  float* out = (float*)d_out;
  float* ws  = (float*)d_ws;
  (void)in_sizes; (void)n_in; (void)out_size; (void)ws_size;

  init_ws_kernel<<<1, 32, 0, stream>>>(ws);
  flatness_kernel<<<N_PTS / 256, 256, 0, stream>>>(scl, ws);
  dim3 agrid(NTILES, CSPLIT);
  align_kernel<<<agrid, 32, 0, stream>>>(pos, rot, ws);
  knn_kernel<<<N_PTS / KNN_WAVES_PER_BLOCK, KNN_WAVES_PER_BLOCK * 32, 0, stream>>>(pos, opa, ws);
  finalize_kernel<<<1, 1, 0, stream>>>(ws, out);
}